// KDDiscriminatorLoss_26010321945360
// MI455X (gfx1250) — compile-verified
//
#include <hip/hip_runtime.h>
#include <math.h>
#include <stdint.h>

typedef float v2f __attribute__((ext_vector_type(2)));
typedef float v8f __attribute__((ext_vector_type(8)));

#define BTOT 4096
#define DK   512
#define BM   128
#define BN   128
#define BK   32
#define LDT  36      // LDS row stride in floats; LDT*4=144 bytes -> 16B-aligned rows
#define NKT  (DK / BK)
#define BUFF ((BM + BN) * LDT)   // one double-buffer slot, in floats
#define P_THRESH 0.7f
#define COS_EPS  1e-6f

// ---------------- reductions ----------------
__device__ inline float wave_reduce(float v) {
#pragma unroll
  for (int off = 16; off > 0; off >>= 1) v += __shfl_xor(v, off, 32);
  return v;
}

__device__ inline float block_reduce_1024(float v, float* smem32) {
  const int lane = threadIdx.x & 31;
  const int w    = threadIdx.x >> 5;
  v = wave_reduce(v);
  __syncthreads();
  if (lane == 0) smem32[w] = v;
  __syncthreads();
  float r = (threadIdx.x < 32) ? smem32[threadIdx.x] : 0.f;
  if (w == 0) {
    r = wave_reduce(r);
    if (lane == 0) smem32[0] = r;
  }
  __syncthreads();
  return smem32[0];
}

// ---------------- 1) AGKD: per-row mean over D ----------------
__global__ __launch_bounds__(256) void agkd_kernel(
    const float* __restrict__ rf, const float* __restrict__ gf,
    const float* __restrict__ rc, const float* __restrict__ fc,
    const float* __restrict__ q, float* __restrict__ agkd) {
  const int row  = blockIdx.x * 8 + (threadIdx.x >> 5);
  const int lane = threadIdx.x & 31;
  const bool use_agg = (q[0] <= P_THRESH);
  const float4* rf4 = (const float4*)(rf + (size_t)row * DK);
  const float4* gf4 = (const float4*)(gf + (size_t)row * DK);
  const float4* rc4 = (const float4*)(rc + (size_t)row * DK);
  const float4* fc4 = (const float4*)(fc + (size_t)row * DK);
  float s = 0.f;
#pragma unroll
  for (int c = lane; c < DK / 4; c += 32) {
    float4 a = rf4[c], g = gf4[c], r = rc4[c], f = fc4[c];
    float kd  = fabsf(r.x - a.x) + fabsf(f.x - g.x)
              + fabsf(r.y - a.y) + fabsf(f.y - g.y)
              + fabsf(r.z - a.z) + fabsf(f.z - g.z)
              + fabsf(r.w - a.w) + fabsf(f.w - g.w);
    float agg = fabsf(r.x - g.x) + fabsf(f.x - a.x)
              + fabsf(r.y - g.y) + fabsf(f.y - a.y)
              + fabsf(r.z - g.z) + fabsf(f.z - a.z)
              + fabsf(r.w - g.w) + fabsf(f.w - a.w);
    s += kd + (use_agg ? agg : 0.f);
  }
  s = wave_reduce(s);
  if (lane == 0) agkd[row] = s * (1.0f / (float)DK);
}

// ---------------- 2) WMMA fp32 GEMM: C = X @ T^T (async-LDS double-buffered) ----------------
__device__ inline void async_stage(const float* __restrict__ A,
                                   const float* __restrict__ T,
                                   float* buf, int rowBase, int colBase,
                                   int kb, int srow, int scol) {
  // Each thread issues 8 global_load_async_to_lds_b128 (ASYNCcnt-tracked).
  // LDS address operand is derived from the real shared pointer so that smem
  // escapes into the asm: the "memory" clobber then keeps the ds_load reads live.
#pragma unroll
  for (int r = 0; r < BM; r += 32) {
    const unsigned lA = (unsigned)(uintptr_t)&buf[(srow + r) * LDT + scol];
    const unsigned gA = (unsigned)((((rowBase + srow + r) * DK) + kb + scol) * 4);
    asm volatile("global_load_async_to_lds_b128 %0, %1, %2"
                 :: "v"(lA), "v"(gA), "s"(A) : "memory");
    const unsigned lB = (unsigned)(uintptr_t)&buf[(BM + srow + r) * LDT + scol];
    const unsigned gB = (unsigned)((((colBase + srow + r) * DK) + kb + scol) * 4);
    asm volatile("global_load_async_to_lds_b128 %0, %1, %2"
                 :: "v"(lB), "v"(gB), "s"(T) : "memory");
  }
}

__global__ __launch_bounds__(256) void wmma_gemm_nt(
    const float* __restrict__ A0, const float* __restrict__ A1,
    const float* __restrict__ T,
    float* __restrict__ C0, float* __restrict__ C1) {
  __shared__ __align__(16) float smem[2 * BUFF];   // 73728 bytes

  const float* __restrict__ A = (blockIdx.z == 0) ? A0 : A1;
  float* __restrict__ C       = (blockIdx.z == 0) ? C0 : C1;

  const int rowBase = blockIdx.y * BM;
  const int colBase = blockIdx.x * BN;

  const int tid   = threadIdx.x;
  const int wave  = tid >> 5;
  const int lane  = tid & 31;
  const int half  = lane >> 4;        // K-pair select: {k,k+1} vs {k+2,k+3}
  const int lrow  = lane & 15;        // matrix row within 16-tile

  const int waveM = (wave >> 2) * 64; // 0 or 64
  const int waveN = (wave & 3) * 32;  // 0,32,64,96

  v8f acc[4][2];
#pragma unroll
  for (int m = 0; m < 4; ++m)
#pragma unroll
    for (int n = 0; n < 2; ++n) acc[m][n] = (v8f)0.f;

  const int srow = tid >> 3;          // 0..31
  const int scol = (tid & 7) * 4;     // 0..28

  // prologue: stage tile 0 into buffer 0
  async_stage(A, T, smem, rowBase, colBase, 0, srow, scol);

  for (int t = 0; t < NKT; ++t) {
    __syncthreads();  // buffer (t+1)&1 no longer being read by any wave
    if (t + 1 < NKT) {
      async_stage(A, T, smem + (((t + 1) & 1) ? BUFF : 0),
                  rowBase, colBase, (t + 1) * BK, srow, scol);
      asm volatile("s_wait_asynccnt 0x8" ::: "memory");  // tile t complete
    } else {
      asm volatile("s_wait_asynccnt 0x0" ::: "memory");
    }
    __syncthreads();  // tile t visible to all waves

    const float* __restrict__ lA = smem + ((t & 1) ? BUFF : 0);
    const float* __restrict__ lB = lA + BM * LDT;
#pragma unroll
    for (int k0 = 0; k0 < BK; k0 += 4) {
      const int kk = k0 + 2 * half;
      v2f a[4], b[2];
#pragma unroll
      for (int m = 0; m < 4; ++m)
        a[m] = *(const v2f*)&lA[(waveM + m * 16 + lrow) * LDT + kk];
#pragma unroll
      for (int n = 0; n < 2; ++n)
        b[n] = *(const v2f*)&lB[(waveN + n * 16 + lrow) * LDT + kk];
#pragma unroll
      for (int m = 0; m < 4; ++m)
#pragma unroll
        for (int n = 0; n < 2; ++n)
          acc[m][n] = __builtin_amdgcn_wmma_f32_16x16x4_f32(
              false, a[m], false, b[n], (short)0, acc[m][n], false, false);
    }
  }

  // store: element (M = r + 8*half, N = lrow) of each 16x16 tile
#pragma unroll
  for (int m = 0; m < 4; ++m) {
#pragma unroll
    for (int n = 0; n < 2; ++n) {
      const int gm0 = rowBase + waveM + m * 16 + 8 * half;
      const int gn  = colBase + waveN + n * 16 + lrow;
#pragma unroll
      for (int r = 0; r < 8; ++r)
        C[(size_t)(gm0 + r) * BTOT + gn] = acc[m][n][r];
    }
  }
}

// ---------------- 3) per-row sum of squares of ct and cs ----------------
__global__ __launch_bounds__(256) void rowsq_kernel(
    const float* __restrict__ ct, const float* __restrict__ cs,
    float* __restrict__ rq_ct, float* __restrict__ rq_cs) {
  const int row  = blockIdx.x * 8 + (threadIdx.x >> 5);
  const int lane = threadIdx.x & 31;
  const float4* c4 = (const float4*)(ct + (size_t)row * BTOT);
  const float4* d4 = (const float4*)(cs + (size_t)row * BTOT);
  float s0 = 0.f, s1 = 0.f;
  for (int c = lane; c < BTOT / 4; c += 32) {
    float4 v = c4[c], u = d4[c];
    s0 += v.x * v.x + v.y * v.y + v.z * v.z + v.w * v.w;
    s1 += u.x * u.x + u.y * u.y + u.z * u.z + u.w * u.w;
  }
  s0 = wave_reduce(s0);
  s1 = wave_reduce(s1);
  if (lane == 0) { rq_ct[row] = s0; rq_cs[row] = s1; }
}

// ---------------- 4) norms / trace / weights ----------------
__global__ __launch_bounds__(1024) void scales_kernel(
    const float* __restrict__ rq_ct, const float* __restrict__ rq_cs,
    float* __restrict__ wvec, float* __restrict__ svec, float* __restrict__ scal) {
  __shared__ float sm[32];
  float s0 = 0.f, s1 = 0.f;
  for (int i = threadIdx.x; i < BTOT; i += 1024) { s0 += rq_ct[i]; s1 += rq_cs[i]; }
  const float nf2_ct = block_reduce_1024(s0, sm);
  const float nf2_cs = block_reduce_1024(s1, sm);
  const float inv_ct = 1.0f / sqrtf(nf2_ct);
  const float inv_cs = 1.0f / sqrtf(nf2_cs);
  float tp = 0.f;
  for (int i = threadIdx.x; i < BTOT; i += 1024) {
    const float r = sqrtf(rq_ct[i]) * inv_ct;           // ||c_t row||
    const float m = fmaxf(r, COS_EPS);
    const float t = r / m;
    tp += t * t;                                        // trace contribution
    wvec[i] = inv_ct / m;                               // cn row scale
    svec[i] = 0.f;                                      // zero column-sum accumulator
  }
  const float trace = block_reduce_1024(tp, sm);
  if (threadIdx.x == 0) { scal[0] = inv_ct; scal[1] = inv_cs; scal[2] = trace; }
}

// ---------------- 5) per-row mean |c_t - c_s| ----------------
__global__ __launch_bounds__(256) void rowmean_kernel(
    const float* __restrict__ ct, const float* __restrict__ cs,
    const float* __restrict__ scal, float* __restrict__ rowmean) {
  const int row  = blockIdx.x * 8 + (threadIdx.x >> 5);
  const int lane = threadIdx.x & 31;
  const float a = scal[0], b = scal[1];
  const float4* c4 = (const float4*)(ct + (size_t)row * BTOT);
  const float4* d4 = (const float4*)(cs + (size_t)row * BTOT);
  float s = 0.f;
  for (int c = lane; c < BTOT / 4; c += 32) {
    float4 v = c4[c], u = d4[c];
    s += fabsf(v.x * a - u.x * b) + fabsf(v.y * a - u.y * b)
       + fabsf(v.z * a - u.z * b) + fabsf(v.w * a - u.w * b);
  }
  s = wave_reduce(s);
  if (lane == 0) rowmean[row] = s * (1.0f / (float)BTOT);
}

// ---------------- 6) column sums: s_j = sum_i ct[i,j] * w_i ----------------
__global__ __launch_bounds__(256) void colsum_kernel(
    const float* __restrict__ ct, const float* __restrict__ wvec,
    float* __restrict__ svec) {
  const int j  = blockIdx.x * 256 + threadIdx.x;
  const int i0 = blockIdx.y * 256;
  float s = 0.f;
#pragma unroll 4
  for (int i = i0; i < i0 + 256; ++i)
    s += ct[(size_t)i * BTOT + j] * wvec[i];
  atomicAdd(&svec[j], s);
}

// ---------------- 7) l_pd = ||s||^2 - trace ----------------
__global__ __launch_bounds__(1024) void lpd_kernel(
    const float* __restrict__ svec, float* __restrict__ scal) {
  __shared__ float sm[32];
  float s = 0.f;
  for (int i = threadIdx.x; i < BTOT; i += 1024) { float v = svec[i]; s += v * v; }
  const float tot = block_reduce_1024(s, sm);
  if (threadIdx.x == 0) scal[3] = tot - scal[2];
}

// ---------------- 8) finalize ----------------
__global__ __launch_bounds__(256) void final_kernel(
    const float* __restrict__ agkd, const float* __restrict__ rowmean,
    const float* __restrict__ scal, float* __restrict__ out) {
  const int i = blockIdx.x * 256 + threadIdx.x;
  out[i] = agkd[i] + scal[3] + rowmean[i];
}

extern "C" void kernel_launch(void* const* d_in, const int* in_sizes, int n_in,
                              void* d_out, int out_size, void* d_ws, size_t ws_size,
                              hipStream_t stream) {
  const float* rf = (const float*)d_in[0];  // real_features
  const float* gf = (const float*)d_in[1];  // gen_features
  const float* rc = (const float*)d_in[2];  // real_clip
  const float* fc = (const float*)d_in[3];  // fake_clip
  const float* tx = (const float*)d_in[4];  // text_features
  const float* q  = (const float*)d_in[5];  // q [1]
  float* out = (float*)d_out;

  float* ws = (float*)d_ws;
  const size_t NN = (size_t)BTOT * BTOT;
  float* ct      = ws;                // [B,B] fake_clip @ text^T
  float* cs      = ws + NN;           // [B,B] gen_features @ text^T
  float* rq_ct   = ws + 2 * NN;
  float* rq_cs   = rq_ct + BTOT;
  float* agkd    = rq_cs + BTOT;
  float* rowmean = agkd + BTOT;
  float* svec    = rowmean + BTOT;
  float* wvec    = svec + BTOT;
  float* scal    = wvec + BTOT;       // [inv_ct, inv_cs, trace, l_pd]

  agkd_kernel<<<BTOT / 8, 256, 0, stream>>>(rf, gf, rc, fc, q, agkd);
  wmma_gemm_nt<<<dim3(BTOT / BN, BTOT / BM, 2), 256, 0, stream>>>(fc, gf, tx, ct, cs);
  rowsq_kernel<<<BTOT / 8, 256, 0, stream>>>(ct, cs, rq_ct, rq_cs);
  scales_kernel<<<1, 1024, 0, stream>>>(rq_ct, rq_cs, wvec, svec, scal);
  rowmean_kernel<<<BTOT / 8, 256, 0, stream>>>(ct, cs, scal, rowmean);
  colsum_kernel<<<dim3(BTOT / 256, BTOT / 256), 256, 0, stream>>>(ct, wvec, svec);
  lpd_kernel<<<1, 1024, 0, stream>>>(svec, scal);
  final_kernel<<<BTOT / 256, 256, 0, stream>>>(agkd, rowmean, scal, out);
}